// SGConv_42923903156363
// MI455X (gfx1250) — compile-verified
//
#include <hip/hip_runtime.h>

namespace {
constexpr int kNodes = 50000;
constexpr int kEdges = 1600000;
constexpr int kD     = 128;
constexpr int kMT    = kNodes / 16;   // 3125 exact 16-row tiles
}

typedef __attribute__((ext_vector_type(2))) float v2f;
typedef __attribute__((ext_vector_type(8))) float v8f;

// ---------------- degree / norm ----------------

__global__ void zero_kernel(float* __restrict__ p, int n) {
  int i = blockIdx.x * blockDim.x + threadIdx.x;
  if (i < n) p[i] = 0.0f;
}

__global__ void degree_kernel(const int* __restrict__ dst, float* __restrict__ degs) {
  int e = blockIdx.x * blockDim.x + threadIdx.x;
  if (e < kEdges) unsafeAtomicAdd(&degs[dst[e]], 1.0f);
}

__global__ void norm_kernel(float* __restrict__ degs_to_norm) {
  int i = blockIdx.x * blockDim.x + threadIdx.x;
  if (i < kNodes) {
    float d = fmaxf(degs_to_norm[i], 1.0f);
    degs_to_norm[i] = rsqrtf(d);   // in place: degs -> norm
  }
}

// ---------------- hop scaling (fused) ----------------

// xs = feat * norm ; acc = 0
__global__ void scale_init_kernel(const float* __restrict__ feat, const float* __restrict__ norm,
                                  float* __restrict__ xs, float* __restrict__ acc) {
  int i = blockIdx.x * blockDim.x + threadIdx.x;
  if (i < kNodes * kD) {
    xs[i]  = feat[i] * norm[i >> 7];
    acc[i] = 0.0f;
  }
}

// xs = acc * norm^2 (post-scale of hop1 fused with pre-scale of hop2); acc = 0
__global__ void rescale_kernel(float* __restrict__ acc, const float* __restrict__ norm,
                               float* __restrict__ xs) {
  int i = blockIdx.x * blockDim.x + threadIdx.x;
  if (i < kNodes * kD) {
    float nm = norm[i >> 7];
    xs[i]  = acc[i] * nm * nm;
    acc[i] = 0.0f;
  }
}

// ---------------- edge scatter-add ----------------
// one wave per edge: lane l moves 16 bytes (float4) of the 512-byte row
__global__ void scatter_kernel(const float* __restrict__ xs, const int* __restrict__ src,
                               const int* __restrict__ dst, float* __restrict__ acc) {
  unsigned t = blockIdx.x * blockDim.x + threadIdx.x;
  int e    = (int)(t >> 5);
  int lane = (int)(t & 31u);
  if (e < kEdges) {
    int s = src[e];
    int d = dst[e];
    const float4 v = *reinterpret_cast<const float4*>(xs + (size_t)s * kD + lane * 4);
    float* base = acc + (size_t)d * kD + lane * 4;
    unsafeAtomicAdd(base + 0, v.x);
    unsafeAtomicAdd(base + 1, v.y);
    unsafeAtomicAdd(base + 2, v.z);
    unsafeAtomicAdd(base + 3, v.w);
  }
}

// ---------------- WMMA f32 GEMM: out = (z * norm) @ W^T + b ----------------
// wave-per-M-tile, 8 N-tiles per wave, V_WMMA_F32_16X16X4_F32, K stepped by 4.
__global__ void __launch_bounds__(256) gemm_wmma_kernel(
    const float* __restrict__ z, const float* __restrict__ norm,
    const float* __restrict__ W, const float* __restrict__ bias,
    float* __restrict__ out) {
  const int wave  = threadIdx.x >> 5;
  const int lane  = threadIdx.x & 31;
  const int mtile = blockIdx.x * 8 + wave;
  if (mtile >= kMT) return;              // wave-uniform branch; active waves keep EXEC all-1s

  const int mrow  = mtile * 16 + (lane & 15);   // A: lane holds row m = lane%16
  const int khalf = lane >> 4;                  // lanes 16..31 cover K+2,K+3
  const int ncol  = lane & 15;                  // B/D: lane holds col n = lane%16
  const float nm  = norm[mrow];                 // fold final norm-scale into A
  const float* __restrict__ arow = z + (size_t)mrow * kD + 2 * khalf;

  v8f acc[8];
#pragma unroll
  for (int t = 0; t < 8; ++t)
    acc[t] = v8f{0.f, 0.f, 0.f, 0.f, 0.f, 0.f, 0.f, 0.f};

#pragma unroll 4
  for (int kb = 0; kb < kD; kb += 4) {
    v2f a = *reinterpret_cast<const v2f*>(arow + kb);   // A[m][kb+2*khalf .. +1]
    a.x *= nm;
    a.y *= nm;
#pragma unroll
    for (int t = 0; t < 8; ++t) {
      // B[k][n] = W[n][k]: contiguous pair from row n of W
      const float* brow = W + (size_t)(t * 16 + ncol) * kD + 2 * khalf + kb;
      v2f b = *reinterpret_cast<const v2f*>(brow);
      acc[t] = __builtin_amdgcn_wmma_f32_16x16x4_f32(
          /*neg_a=*/false, a, /*neg_b=*/false, b,
          /*c_mod=*/(short)0, acc[t], /*reuse_a=*/false, /*reuse_b=*/false);
    }
  }

  // D layout: VGPR r, lanes 0-15 -> row r, lanes 16-31 -> row r+8; col = lane%16
#pragma unroll
  for (int t = 0; t < 8; ++t) {
    const int col  = t * 16 + ncol;
    const float bv = bias[col];
    float* ocol = out + (size_t)(mtile * 16 + 8 * khalf) * kD + col;
#pragma unroll
    for (int r = 0; r < 8; ++r) {
      ocol[(size_t)r * kD] = acc[t][r] + bv;
    }
  }
}

// ---------------- host ----------------

extern "C" void kernel_launch(void* const* d_in, const int* in_sizes, int n_in,
                              void* d_out, int out_size, void* d_ws, size_t ws_size,
                              hipStream_t stream) {
  (void)in_sizes; (void)n_in; (void)out_size; (void)ws_size;

  const float* feat = (const float*)d_in[0];
  const int*   src  = (const int*)d_in[1];
  const int*   dst  = (const int*)d_in[2];
  const float* W    = (const float*)d_in[3];
  const float* bias = (const float*)d_in[4];
  float*       out  = (float*)d_out;

  // workspace: norm (50000 f) | bufA (50000x128 f) | bufB (50000x128 f)  ~= 51.4 MB
  char* ws = (char*)d_ws;
  float* norm = (float*)ws;
  size_t off  = (((size_t)kNodes * 4) + 255) & ~(size_t)255;
  float* bufA = (float*)(ws + off);
  float* bufB = (float*)(ws + off + (size_t)kNodes * kD * sizeof(float));

  const int B = 256;
  const int nodeBlocks = (kNodes + B - 1) / B;
  const int featBlocks = (kNodes * kD + B - 1) / B;
  const int edgeBlocks = (kEdges + B - 1) / B;
  const long long sthreads = (long long)kEdges * 32;
  const int scatBlocks = (int)((sthreads + B - 1) / B);   // 200000

  zero_kernel<<<nodeBlocks, B, 0, stream>>>(norm, kNodes);
  degree_kernel<<<edgeBlocks, B, 0, stream>>>(dst, norm);
  norm_kernel<<<nodeBlocks, B, 0, stream>>>(norm);

  // hop 1
  scale_init_kernel<<<featBlocks, B, 0, stream>>>(feat, norm, bufA, bufB);
  scatter_kernel<<<scatBlocks, B, 0, stream>>>(bufA, src, dst, bufB);

  // hop 2 (post-scale hop1 * pre-scale hop2 fused as norm^2)
  rescale_kernel<<<featBlocks, B, 0, stream>>>(bufB, norm, bufA);
  scatter_kernel<<<scatBlocks, B, 0, stream>>>(bufA, src, dst, bufB);

  // projection with final norm folded into A
  gemm_wmma_kernel<<<(kMT + 7) / 8, B, 0, stream>>>(bufB, norm, W, bias, out);
}